// PWLNNFcn_53171695125377
// MI455X (gfx1250) — compile-verified
//
#include <hip/hip_runtime.h>

typedef __attribute__((ext_vector_type(2))) float v2f;
typedef __attribute__((ext_vector_type(4))) float v4f;
typedef __attribute__((ext_vector_type(8))) float v8f;

#define D_IN   32
#define D_OUT  32
#define K_NN   4
#define WAVES_PER_BLOCK 8
#define TILE_M 16
#define TILE_N 16
#define LDS_STRIDE 17   // 17 coprime with 64 banks -> conflict-free row reads

// Fused kernel:
//  Phase 1: per-wave 16-sample tile, loop over all center tiles:
//           v_wmma_f32_16x16x4_f32 chain (K=32) for x @ ctrs^T, score = |c|^2 - 2*dot
//           (|x|^2 dropped: row-constant, cannot change per-row top-k order),
//           running top-4 (smallest score, lowest-index tie-break) per sample.
//  Phase 2: gather k=4 local linear models, y = sum_k (x - c_k) . W_k + off_k.
__global__ __launch_bounds__(256) void pwlnn_fused_kernel(
    const float* __restrict__ x, const float* __restrict__ ctrs,
    const float* __restrict__ wts, const float* __restrict__ offs,
    float* __restrict__ out, int n_fcns)
{
  __shared__ float lds_score[WAVES_PER_BLOCK][TILE_M * LDS_STRIDE];
  __shared__ int   lds_idx[WAVES_PER_BLOCK][TILE_M * K_NN];

  const int lane = threadIdx.x & 31;
  const int wave = threadIdx.x >> 5;
  const int nloc = lane & 15;   // M index for A-frag / N index for B & C frags
  const int h    = lane >> 4;   // lane half selects K pairs (ISA f32 WMMA layout)
  const int m0   = (blockIdx.x * WAVES_PER_BLOCK + wave) * TILE_M;

  // ---- A fragments: x tile (16 samples x 32 dims), f32 16x16x4 A layout:
  //      lane L holds A[M = L%16, K = 4*s + 2*(L/16) + {0,1}] in frag s.
  v2f afrag[8];
#pragma unroll
  for (int s = 0; s < 8; ++s)
    afrag[s] = *(const v2f*)(x + (size_t)(m0 + nloc) * D_IN + 4 * s + 2 * h);

  // running top-4 (sorted ascending) for sample row 'lane' (lanes 0..15 only)
  float s0 = 3.4e38f, s1 = 3.4e38f, s2 = 3.4e38f, s3 = 3.4e38f;
  int   i0 = 0, i1 = 0, i2 = 0, i3 = 0;

  const int ntiles = n_fcns >> 4;
  for (int nt = 0; nt < ntiles; ++nt) {
    const int n0 = nt << 4;
    const float* crow = ctrs + (size_t)(n0 + nloc) * D_IN;

    // B fragments: B[K = 4*s + 2h + {0,1}, N = lane%16]; also |c|^2 partial.
    v2f bfrag[8];
    float cpart = 0.0f;
#pragma unroll
    for (int s = 0; s < 8; ++s) {
      bfrag[s] = *(const v2f*)(crow + 4 * s + 2 * h);
      cpart += bfrag[s].x * bfrag[s].x + bfrag[s].y * bfrag[s].y;
    }
    // lanes L and L^16 share the same center row and jointly cover K
    const float cn = cpart + __shfl_xor(cpart, 16, 32);

    // prefetch next center tile into cache (global_prefetch_b8)
    __builtin_prefetch(crow + TILE_N * D_IN, 0, 0);

    // chained f32 WMMA accumulation over K = 32
    v8f acc = {0.f, 0.f, 0.f, 0.f, 0.f, 0.f, 0.f, 0.f};
#pragma unroll
    for (int s = 0; s < 8; ++s)
      acc = __builtin_amdgcn_wmma_f32_16x16x4_f32(
          /*neg_a=*/false, afrag[s], /*neg_b=*/false, bfrag[s],
          /*c_mod=*/(short)0, acc, /*reuse_a=*/false, /*reuse_b=*/false);

    // C layout: VGPR j holds (M = j + 8*h, N = lane%16). Stage scores in LDS.
#pragma unroll
    for (int j = 0; j < 8; ++j)
      lds_score[wave][(j + 8 * h) * LDS_STRIDE + nloc] = cn - 2.0f * acc[j];

    __syncthreads();

    if (lane < 16) {
      // strict '<' => equal scores keep the earlier (lower) center index,
      // matching jax.lax.top_k tie behavior.
#pragma unroll
      for (int n2 = 0; n2 < 16; ++n2) {
        const float v = lds_score[wave][lane * LDS_STRIDE + n2];
        const int col = n0 + n2;
        if (v < s3) {
          if (v < s2) {
            s3 = s2; i3 = i2;
            if (v < s1) {
              s2 = s1; i2 = i1;
              if (v < s0) { s1 = s0; i1 = i0; s0 = v; i0 = col; }
              else        { s1 = v;  i1 = col; }
            } else { s2 = v; i2 = col; }
          } else { s3 = v; i3 = col; }
        }
      }
    }
    __syncthreads();
  }

  // publish per-sample neighbor indices for the whole wave
  if (lane < 16) {
    lds_idx[wave][lane * K_NN + 0] = i0;
    lds_idx[wave][lane * K_NN + 1] = i1;
    lds_idx[wave][lane * K_NN + 2] = i2;
    lds_idx[wave][lane * K_NN + 3] = i3;
  }
  __syncthreads();

  // ---- Phase 2: 2 lanes per sample (h splits the 32 outputs into 2x16)
  const int m  = m0 + nloc;
  const int e0 = h * 16;

  v4f xr[8];
#pragma unroll
  for (int q = 0; q < 8; ++q)
    xr[q] = *(const v4f*)(x + (size_t)m * D_IN + 4 * q);

  v4f acc4[4] = {{0.f,0.f,0.f,0.f},{0.f,0.f,0.f,0.f},
                 {0.f,0.f,0.f,0.f},{0.f,0.f,0.f,0.f}};

#pragma unroll
  for (int j = 0; j < K_NN; ++j) {
    const int ci = lds_idx[wave][nloc * K_NN + j];
    const float* crow = ctrs + (size_t)ci * D_IN;
    const float* orow = offs + (size_t)ci * D_OUT + e0;
    const float* wbase = wts + (size_t)ci * D_IN * D_OUT + e0;

#pragma unroll
    for (int q = 0; q < 4; ++q)
      acc4[q] += *(const v4f*)(orow + 4 * q);

    v4f t4[8];
#pragma unroll
    for (int q = 0; q < 8; ++q)
      t4[q] = xr[q] - *(const v4f*)(crow + 4 * q);

#pragma unroll
    for (int d = 0; d < D_IN; ++d) {
      const float t = t4[d >> 2][d & 3];
      const v4f* w = (const v4f*)(wbase + (size_t)d * D_OUT);
      acc4[0] += w[0] * t;
      acc4[1] += w[1] * t;
      acc4[2] += w[2] * t;
      acc4[3] += w[3] * t;
    }
  }

  float* yrow = out + (size_t)m * D_OUT + e0;
#pragma unroll
  for (int q = 0; q < 4; ++q)
    *(v4f*)(yrow + 4 * q) = acc4[q];
}

extern "C" void kernel_launch(void* const* d_in, const int* in_sizes, int n_in,
                              void* d_out, int out_size, void* d_ws, size_t ws_size,
                              hipStream_t stream) {
  const float* x    = (const float*)d_in[0];
  const float* ctrs = (const float*)d_in[1];
  const float* wts  = (const float*)d_in[2];
  const float* offs = (const float*)d_in[3];
  float* out = (float*)d_out;

  const int n_smps = in_sizes[0] / D_IN;   // 16384
  const int n_fcns = in_sizes[1] / D_IN;   // 2048

  const int blocks = n_smps / (WAVES_PER_BLOCK * TILE_M);  // 128
  pwlnn_fused_kernel<<<blocks, 256, 0, stream>>>(x, ctrs, wts, offs, out, n_fcns);
}